// RewardFunctionRegret_32856499814607
// MI455X (gfx1250) — compile-verified
//
#include <hip/hip_runtime.h>
#include <hip/hip_bf16.h>

typedef float v2f __attribute__((ext_vector_type(2)));
typedef float v8f __attribute__((ext_vector_type(8)));

// Problem constants from the reference
#define NB    32768   // batch
#define NP    128     // policies
#define NX    30
#define NY    30
#define NCELL (NX*NY) // 900
#define NF    6
#define INV_T 1000.0f

// ---------------------------------------------------------------------------
// Phase 1: build sval table: table[cell*128 + p] = dot(succ_feats[p,cell,:6], w)
// One wave -> one 16-row tile (16 consecutive p for a fixed cell), computed
// with two V_WMMA_F32_16X16X4_F32 steps (K chunks f0..f3 and f4,f5,0,0).
// B is w broadcast across all 16 N columns; column N=0 of D is extracted by
// lanes 0 (M=0..7) and 16 (M=8..15) and stored contiguously.
// Grid: 900 blocks x 256 threads (8 waves/block), cell = blockIdx.x.
// ---------------------------------------------------------------------------
__global__ __launch_bounds__(256)
void sval_table_wmma_kernel(const float* __restrict__ sf,   // (P, X, Y, F)
                            const float* __restrict__ w,    // (F,)
                            float* __restrict__ table)      // (NCELL, NP)
{
    const int lane   = threadIdx.x & 31;
    const int waveId = threadIdx.x >> 5;
    const int cell   = blockIdx.x;        // 0..899
    const int p0     = waveId << 4;       // 0,16,...,112
    const int m      = lane & 15;         // row within tile
    const int hi     = lane >> 4;         // 0: K=0,1  1: K=2,3
    const int p      = p0 + m;

    const long base = ((long)p * NCELL + cell) * NF;

    // A chunk 0: K = (hi?2:0), (hi?3:1)  -> features f0..f3
    v2f a0, a1, b0, b1;
    a0.x = sf[base + (hi ? 2 : 0)];
    a0.y = sf[base + (hi ? 3 : 1)];
    // A chunk 1: K=0,1 -> f4,f5 ; K=2,3 -> zero pad
    a1.x = hi ? 0.0f : sf[base + 4];
    a1.y = hi ? 0.0f : sf[base + 5];

    // B: broadcast w across all N columns (same K split as A halves)
    b0.x = w[hi ? 2 : 0];
    b0.y = w[hi ? 3 : 1];
    b1.x = hi ? 0.0f : w[4];
    b1.y = hi ? 0.0f : w[5];

    v8f c = {};
    c = __builtin_amdgcn_wmma_f32_16x16x4_f32(false, a0, false, b0,
                                              (short)0, c, false, false);
    c = __builtin_amdgcn_wmma_f32_16x16x4_f32(false, a1, false, b1,
                                              (short)0, c, false, false);

    // D column N=0: lane 0 holds M=0..7 in c[0..7]; lane 16 holds M=8..15.
    if (m == 0) {
        float* dst = table + (long)cell * NP + p0 + hi * 8;
#pragma unroll
        for (int j = 0; j < 8; ++j) dst[j] = c[j];
    }
}

// ---------------------------------------------------------------------------
// Phase 2: one wave per batch element b.
// 4 softmax-weighted reductions over 128 table values (t={0,1} x {ss,es}),
// each lane owning 4 values; cross-lane max/sum via __shfl_xor (wave32).
// ---------------------------------------------------------------------------
__global__ __launch_bounds__(256)
void regret_softmax_kernel(const float* __restrict__ phi,   // (B, 2, 10)
                           const float* __restrict__ table, // (NCELL, NP)
                           const float* __restrict__ w,     // (F,)
                           float* __restrict__ out)         // (B, 2, 1)
{
    const int lane = threadIdx.x & 31;
    const int b    = blockIdx.x * 8 + (threadIdx.x >> 5);

    const float* ph = phi + (long)b * 20;

    float pr[2];
    int   idxs[4];   // [t*2+0]=ss, [t*2+1]=es
#pragma unroll
    for (int t = 0; t < 2; ++t) {
        float s = 0.0f;
#pragma unroll
        for (int f = 0; f < NF; ++f) s += ph[t * 10 + f] * w[f];
        pr[t] = s;
        idxs[t * 2 + 0] = (int)ph[t * 10 + 6] * NY + (int)ph[t * 10 + 7];
        idxs[t * 2 + 1] = (int)ph[t * 10 + 8] * NY + (int)ph[t * 10 + 9];
    }

    float vred[4];
#pragma unroll
    for (int r = 0; r < 4; ++r) {
        const float* col = table + (long)idxs[r] * NP;
        float v0 = col[lane      ];
        float v1 = col[lane + 32 ];
        float v2 = col[lane + 64 ];
        float v3 = col[lane + 96 ];

        float mx = fmaxf(fmaxf(v0, v1), fmaxf(v2, v3));
#pragma unroll
        for (int off = 16; off > 0; off >>= 1)
            mx = fmaxf(mx, __shfl_xor(mx, off, 32));

        float e0 = __expf((v0 - mx) * INV_T);
        float e1 = __expf((v1 - mx) * INV_T);
        float e2 = __expf((v2 - mx) * INV_T);
        float e3 = __expf((v3 - mx) * INV_T);
        float s  = e0 + e1 + e2 + e3;
        float ws = e0 * v0 + e1 * v1 + e2 * v2 + e3 * v3;
#pragma unroll
        for (int off = 16; off > 0; off >>= 1) {
            s  += __shfl_xor(s,  off, 32);
            ws += __shfl_xor(ws, off, 32);
        }
        vred[r] = ws / s;
    }

    if (lane == 0) {
        float d0 = pr[0] + vred[1] - vred[0];  // es - ss, t=0
        float d1 = pr[1] + vred[3] - vred[2];  // es - ss, t=1
        float d  = d0 - d1;
        out[(long)b * 2 + 0] = 1.0f / (1.0f + __expf(-d));  // left_pred
        out[(long)b * 2 + 1] = 1.0f / (1.0f + __expf( d));  // right_pred
    }
}

extern "C" void kernel_launch(void* const* d_in, const int* in_sizes, int n_in,
                              void* d_out, int out_size, void* d_ws, size_t ws_size,
                              hipStream_t stream)
{
    const float* phi = (const float*)d_in[0];  // 32768*2*10
    const float* sf  = (const float*)d_in[1];  // 128*30*30*6
    const float* w   = (const float*)d_in[2];  // 6
    float* out   = (float*)d_out;              // 32768*2
    float* table = (float*)d_ws;               // 900*128 floats = 450 KB

    // Phase 1: WMMA table build. 900 blocks * 8 waves = 7200 tiles of 16 rows.
    sval_table_wmma_kernel<<<NCELL, 256, 0, stream>>>(sf, w, table);

    // Phase 2: one wave per b. 32768 / 8 waves-per-block = 4096 blocks.
    regret_softmax_kernel<<<NB / 8, 256, 0, stream>>>(phi, table, w, out);
}